// locally_masked_conv2d_35983236006479
// MI455X (gfx1250) — compile-verified
//
#include <hip/hip_runtime.h>

typedef __attribute__((ext_vector_type(2))) float v2f;
typedef __attribute__((ext_vector_type(4))) float v4f;
typedef __attribute__((ext_vector_type(8))) float v8f;

#define B_    16
#define CIN_  128
#define COUT_ 128
#define HW_   64
#define L_    4096
#define KT_   9

#define APACK_F4 (KT_ * 64 * 64)   // [k][c2(64)][g(64)] float4
#define MPACK_F4 (6 * 64)          // [c2(6)][g(64)] float4 (K padded 9->12)

// xs: [row(3)][c2(64)][col(66)][par(2)]  raw x with halo cols 0 and 65 zeroed
#define XS_FLOATS (3 * 64 * 66 * 2)
// ms: [c2(6)][col(64)][par(2)]           mask tile, K padded 9->12
#define MS_FLOATS (6 * 64 * 2)
#define LDS_BYTES ((XS_FLOATS + MS_FLOATS) * 4)

// ---------------- weight / mask_weight prepack into A-fragment float4 layout ---
// g = q*16 + j encodes (o_lo = 32q + j, o_hi = o_lo + 16); element:
// ( W[o_lo][2c2][k], W[o_lo][2c2+1][k], W[o_hi][2c2][k], W[o_hi][2c2+1][k] )
__global__ __launch_bounds__(256) void lmconv_prepack(
    const float* __restrict__ weight,   // [COUT][CIN][3][3]
    const float* __restrict__ mw,       // [COUT][3][3]
    float* __restrict__ ws) {
  int idx = blockIdx.x * blockDim.x + threadIdx.x;
  v4f* apack = (v4f*)ws;
  v4f* mpack = (v4f*)ws + APACK_F4;
  if (idx < APACK_F4) {
    int k   = idx >> 12;            // /4096
    int rem = idx & 4095;
    int c2  = rem >> 6;
    int g   = rem & 63;
    int olo = ((g >> 4) << 5) + (g & 15);
    int ohi = olo + 16;
    int c0  = c2 << 1;
    v4f v;
    v.x = weight[(olo * CIN_ + c0) * 9 + k];
    v.y = weight[(olo * CIN_ + c0 + 1) * 9 + k];
    v.z = weight[(ohi * CIN_ + c0) * 9 + k];
    v.w = weight[(ohi * CIN_ + c0 + 1) * 9 + k];
    apack[idx] = v;
  } else if (idx < APACK_F4 + MPACK_F4) {
    int r   = idx - APACK_F4;
    int c2  = r >> 6;
    int g   = r & 63;
    int olo = ((g >> 4) << 5) + (g & 15);
    int ohi = olo + 16;
    int c0  = c2 << 1;
    v4f v;
    v.x = (c0     < 9) ? mw[olo * 9 + c0]     : 0.0f;
    v.y = (c0 + 1 < 9) ? mw[olo * 9 + c0 + 1] : 0.0f;
    v.z = (c0     < 9) ? mw[ohi * 9 + c0]     : 0.0f;
    v.w = (c0 + 1 < 9) ? mw[ohi * 9 + c0 + 1] : 0.0f;
    mpack[r] = v;
  }
}

// ---------------- main: WG = (batch, image row) -> 128 COUT x 64 pixels ---------
// 8 waves = 4(M) x 2(N); wave tile 32x32; per K-chunk: 1 b128 A + 2 b64 B -> 4 WMMA
__global__ __launch_bounds__(256) void lmconv_main(
    const float* __restrict__ x,      // [B][CIN][64][64]
    const float* __restrict__ mask,   // [9][4096]
    const float* __restrict__ bias,   // [COUT]
    const float* __restrict__ ws,
    float* __restrict__ out) {        // [B][COUT][4096]
  extern __shared__ float smem[];
  float* xs = smem;               // [3][64][66][2]
  float* ms = smem + XS_FLOATS;   // [6][64][2]

  const int tid  = threadIdx.x;
  const int lane = tid & 31;
  const int wave = tid >> 5;
  const int b    = blockIdx.y;
  const int h0   = blockIdx.x;
  const int l0   = h0 << 6;

  const int j    = lane & 15;       // N column within subtile == M row within A frag
  const int half = lane >> 4;       // K-pair selector
  const int q    = wave >> 1;       // M group: m0 = 32*q
  const int m0   = q << 5;
  const int nb   = (wave & 1) << 5; // N base: 0 or 32

  const v4f* apack = (const v4f*)ws;
  const v4f* mpack = (const v4f*)ws + APACK_F4;

  // ---- stage mask tile (pair layout, K padded to 12) : 768 floats ----
#pragma unroll
  for (int i = 0; i < 3; ++i) {
    int e   = i * 256 + tid;
    int c2  = e >> 7;
    int col = (e >> 1) & 63;
    int par = e & 1;
    int c   = (c2 << 1) + par;
    ms[e] = (c < 9) ? mask[c * L_ + l0 + col] : 0.0f;
  }
  // ---- stage raw x rows h0-1, h0, h0+1 (no mask, no shift): 24576 floats ----
#pragma unroll
  for (int i = 0; i < 24; ++i) {
    int e   = i * 256 + tid;        // float4 group id over [3][128][16]
    int r   = e >> 11;
    int rem = e & 2047;
    int c   = rem >> 4;
    int w4  = (rem & 15) << 2;
    int h   = h0 + r - 1;
    v4f v = {};
    if ((unsigned)h < 64u)          // workgroup-uniform branch
      v = *(const v4f*)&x[(b * CIN_ + c) * L_ + (h << 6) + w4];
    int base = (((r << 6) + (c >> 1)) * 66 + (w4 + 1)) * 2 + (c & 1);
    xs[base + 0] = v.x;
    xs[base + 2] = v.y;
    xs[base + 4] = v.z;
    xs[base + 6] = v.w;
  }
  // ---- zero halo columns cl=0 and cl=65 : 768 writes ----
#pragma unroll
  for (int i = 0; i < 3; ++i) {
    int e    = i * 256 + tid;       // over [3][64][2][2]
    int r    = e >> 8;
    int rem  = e & 255;
    int c2   = rem >> 2;
    int par  = (rem >> 1) & 1;
    int cl   = (rem & 1) ? 65 : 0;
    xs[(((r << 6) + c2) * 66 + cl) * 2 + par] = 0.0f;
  }
  __syncthreads();

  v8f acc[2][2] = {};

  // ---- mask_weight @ mask term folded as K=12 mini-GEMM (12 WMMAs) ----
#pragma unroll
  for (int kk = 0; kk < 12; kk += 4) {
    int c2 = (kk >> 1) + half;
    v4f a4 = mpack[c2 * 64 + (q << 4) + j];
    v2f aLo = {a4.x, a4.y};
    v2f aHi = {a4.z, a4.w};
#pragma unroll
    for (int ni = 0; ni < 2; ++ni) {
      v2f bb = *(const v2f*)&ms[c2 * 128 + (nb + ni * 16 + j) * 2];
      acc[0][ni] = __builtin_amdgcn_wmma_f32_16x16x4_f32(false, aLo, false, bb,
                                                         (short)0, acc[0][ni], false, false);
      acc[1][ni] = __builtin_amdgcn_wmma_f32_16x16x4_f32(false, aHi, false, bb,
                                                         (short)0, acc[1][ni], false, false);
    }
  }

  // ---- 9 taps: per-tap GEMM on raw x, then fold mask_k(col) with 32 FMAs ----
  for (int k = 0; k < KT_; ++k) {
    const int r  = k / 3;           // row select (dy = r-1)
    const int dx = k % 3 - 1;

    float mval[2];
#pragma unroll
    for (int ni = 0; ni < 2; ++ni)
      mval[ni] = ms[(k >> 1) * 128 + (nb + ni * 16 + j) * 2 + (k & 1)];

    v8f tp[2][2] = {};
#pragma unroll
    for (int cc = 0; cc < CIN_; cc += 4) {
      int c2 = (cc >> 1) + half;
      v4f a4 = apack[(k * 64 + c2) * 64 + (q << 4) + j];   // one b128 -> 2 A frags
      v2f aLo = {a4.x, a4.y};
      v2f aHi = {a4.z, a4.w};
#pragma unroll
      for (int ni = 0; ni < 2; ++ni) {
        int cl = nb + ni * 16 + j + dx + 1;                // halo-shifted column
        v2f bb = *(const v2f*)&xs[(((r << 6) + c2) * 66 + cl) * 2];
        tp[0][ni] = __builtin_amdgcn_wmma_f32_16x16x4_f32(false, aLo, false, bb,
                                                          (short)0, tp[0][ni], false, false);
        tp[1][ni] = __builtin_amdgcn_wmma_f32_16x16x4_f32(false, aHi, false, bb,
                                                          (short)0, tp[1][ni], false, false);
      }
    }
#pragma unroll
    for (int mi = 0; mi < 2; ++mi)
#pragma unroll
      for (int ni = 0; ni < 2; ++ni)
#pragma unroll
        for (int t = 0; t < 8; ++t)
          acc[mi][ni][t] += mval[ni] * tp[mi][ni][t];
  }

  // ---- epilogue: bias + coalesced stores ----
#pragma unroll
  for (int mi = 0; mi < 2; ++mi) {
#pragma unroll
    for (int ni = 0; ni < 2; ++ni) {
      int n = nb + ni * 16 + j;
#pragma unroll
      for (int rr = 0; rr < 8; ++rr) {
        int o = m0 + mi * 16 + (half << 3) + rr;
        out[(b * COUT_ + o) * L_ + l0 + n] = acc[mi][ni][rr] + bias[o];
      }
    }
  }
}

extern "C" void kernel_launch(void* const* d_in, const int* in_sizes, int n_in,
                              void* d_out, int out_size, void* d_ws, size_t ws_size,
                              hipStream_t stream) {
  const float* x      = (const float*)d_in[0];
  const float* mask   = (const float*)d_in[1];
  const float* weight = (const float*)d_in[2];
  const float* mw     = (const float*)d_in[3];
  const float* bias   = (const float*)d_in[4];
  float* ws  = (float*)d_ws;
  float* out = (float*)d_out;

  int pre_total = APACK_F4 + MPACK_F4;
  lmconv_prepack<<<(pre_total + 255) / 256, 256, 0, stream>>>(weight, mw, ws);

  dim3 grid(HW_, B_);   // 64 image rows x 16 batches = 1024 workgroups
  lmconv_main<<<grid, 256, LDS_BYTES, stream>>>(x, mask, bias, ws, out);
}